// GAT_83854941487736
// MI455X (gfx1250) — compile-verified
//
#include <hip/hip_runtime.h>
#include <hip/hip_bf16.h>

#define NN      100000        // nodes
#define NE      1600000       // edges (without self loops)
#define EN      (NE + NN)     // edges + self loops
#define F_INN   128
#define HIDD    64
#define HEADS   3
#define OUT_C   2
#define NEG_SLOPE 0.2f
#define HC1     (HEADS * HIDD)    // 192
#define HC2     (HEADS * OUT_C)   // 6

typedef float v2f __attribute__((ext_vector_type(2)));
typedef float v8f __attribute__((ext_vector_type(8)));

// ---------- order-preserving f32 <-> u32 key (for atomicMax-based segment max) ----------
__device__ __forceinline__ unsigned fkey(float f) {
    unsigned u = __float_as_uint(f);
    return (u & 0x80000000u) ? ~u : (u | 0x80000000u);
}
__device__ __forceinline__ float funkey(unsigned k) {
    return __uint_as_float((k & 0x80000000u) ? (k & 0x7FFFFFFFu) : ~k);
}

__device__ __forceinline__ void edge_sd(const int* __restrict__ ei, long long e,
                                        int& s, int& d) {
    if (e < NE) { s = ei[e]; d = ei[NE + e]; }
    else        { s = d = (int)(e - NE); }     // appended self loops
}

// ---------------------------------------------------------------------------
// FP32 WMMA GEMM: C[M,Nc] = (A[M,K] (+Abias[k])) @ B[K,Nc] (+Cbias[col]), opt ReLU.
// Block = 8 waves, all sharing one 16-column B panel staged (transposed, padded)
// in LDS -> per-step B fetch is one conflict-free ds_load_b64.
// K, ABIAS, RELU are compile-time -> branch-free, unrolled WMMA inner loop.
// Out-of-range waves clamp their A row (EXEC stays all-ones for WMMA/barrier)
// and only skip the final store.
// ---------------------------------------------------------------------------
template<int K, bool ABIAS, bool RELU>
__global__ void __launch_bounds__(256)
wmma_gemm_f32(const float* __restrict__ A, const float* __restrict__ Abias,
              const float* __restrict__ B, const float* __restrict__ Cbias,
              float* __restrict__ C, int M, int Nc)
{
    constexpr int KP = K + 4;                       // pad to avoid LDS bank conflicts
    __shared__ float Blds[16 * KP];

    const int col0 = blockIdx.y << 4;
    // cooperative stage of B panel: Blds[c][k] = B[k][col0+c]
    for (int idx = threadIdx.x; idx < 16 * K; idx += 256) {
        const int k = idx >> 4, c = idx & 15;
        Blds[c * KP + k] = B[(long long)k * Nc + col0 + c];
    }
    __syncthreads();

    const int w    = threadIdx.x >> 5;
    const int lane = threadIdx.x & 31;
    const int mt   = blockIdx.x * 8 + w;
    const int tiles_m = M >> 4;                     // M % 16 == 0
    const bool active = (mt < tiles_m);

    const int rbase = active ? (mt << 4) : (M - 16);
    const int row   = rbase + (lane & 15);
    const int khi   = (lane >> 4) << 1;             // lanes 0-15: K,K+1 ; lanes 16-31: K+2,K+3
    const float* __restrict__ Arow = A + (long long)row * K;
    const float* __restrict__ Bl   = &Blds[(lane & 15) * KP];

    v8f acc = {};
#pragma unroll 4
    for (int k0 = 0; k0 < K; k0 += 4) {
        const int ka = k0 + khi;
        v2f a = *(const v2f*)(Arow + ka);
        if (ABIAS) { a.x += Abias[ka]; a.y += Abias[ka + 1]; }
        const v2f b = *(const v2f*)(Bl + ka);
        acc = __builtin_amdgcn_wmma_f32_16x16x4_f32(
                  false, a, false, b, (short)0, acc, false, false);
    }

    if (active) {
        const int col = col0 + (lane & 15);
        const float cb = Cbias ? Cbias[col] : 0.0f;
#pragma unroll
        for (int i = 0; i < 8; ++i) {
            const int m = (mt << 4) + ((lane < 16) ? i : (i + 8));
            float v = acc[i] + cb;
            if (RELU) v = fmaxf(v, 0.0f);
            C[(long long)m * Nc + col] = v;
        }
    }
}

// ---------------------------------------------------------------------------
// Layer-1 attention logits: a_s[n,h] = <h1[n,h,:], att_src[h,:]>, same for dst
// ---------------------------------------------------------------------------
__global__ void att_logits1(const float* __restrict__ h1,
                            const float* __restrict__ att_s,
                            const float* __restrict__ att_d,
                            float* __restrict__ a_s, float* __restrict__ a_d)
{
    const int t = blockIdx.x * blockDim.x + threadIdx.x;
    if (t >= NN * HEADS) return;
    const int n = t / HEADS, hh = t % HEADS;
    const float* hp = h1 + (long long)n * HC1 + hh * HIDD;
    float s = 0.f, d = 0.f;
#pragma unroll 8
    for (int c = 0; c < HIDD; ++c) {
        const float v = hp[c];
        s += v * att_s[hh * HIDD + c];
        d += v * att_d[hh * HIDD + c];
    }
    a_s[t] = s; a_d[t] = d;
}

// ---------------------------------------------------------------------------
// Edge-parallel segment max (softmax stabilization), L2-resident atomics
// ---------------------------------------------------------------------------
__global__ void edge_max(const int* __restrict__ ei, const float* __restrict__ a_s,
                         const float* __restrict__ a_d, unsigned* __restrict__ emax)
{
    const long long t = (long long)blockIdx.x * blockDim.x + threadIdx.x;
    if (t >= (long long)EN * HEADS) return;
    const long long e = t / HEADS; const int hh = (int)(t % HEADS);
    int s, d; edge_sd(ei, e, s, d);
    float el = a_s[s * HEADS + hh] + a_d[d * HEADS + hh];
    el = (el > 0.f) ? el : NEG_SLOPE * el;
    atomicMax(&emax[d * HEADS + hh], fkey(el));
}

// ---------------------------------------------------------------------------
// Layer-1 aggregation, fused with denominator accumulation:
// aggregate with UNNORMALIZED weight ex = exp(el - max); lane 0 also adds ex
// into den[dst,h]. Normalization happens once per node in norm1.
// One wave32 per (edge, head); each lane moves 2 of 64 channels.
// ---------------------------------------------------------------------------
__global__ void __launch_bounds__(256)
edge_agg1(const int* __restrict__ ei, const float* __restrict__ h1,
          const float* __restrict__ a_s, const float* __restrict__ a_d,
          const unsigned* __restrict__ emax, float* __restrict__ den,
          float* __restrict__ agg)
{
    const long long wid = ((long long)blockIdx.x * blockDim.x + threadIdx.x) >> 5;
    const int lane = threadIdx.x & 31;
    if (wid >= (long long)EN * HEADS) return;
    const long long e = wid / HEADS; const int hh = (int)(wid % HEADS);
    int s, d; edge_sd(ei, e, s, d);
    float el = a_s[s * HEADS + hh] + a_d[d * HEADS + hh];
    el = (el > 0.f) ? el : NEG_SLOPE * el;
    const float ex = __expf(el - funkey(emax[d * HEADS + hh]));
    if (lane == 0) atomicAdd(&den[d * HEADS + hh], ex);
    const float* hp = h1  + (long long)s * HC1 + hh * HIDD;
    float*       op = agg + (long long)d * HC1 + hh * HIDD;
    atomicAdd(&op[lane],      hp[lane]      * ex);
    atomicAdd(&op[lane + 32], hp[lane + 32] * ex);
}

// Per-node normalization of the layer-1 aggregate
__global__ void norm1(float* __restrict__ agg, const float* __restrict__ den)
{
    const int t = blockIdx.x * blockDim.x + threadIdx.x;
    if (t >= NN * HC1) return;
    const int n = t / HC1, hh = (t % HC1) / HIDD;
    agg[t] /= den[n * HEADS + hh];
}

// ---------------------------------------------------------------------------
// Layer-2 node work: h2 = hfc @ W2 (64x6, tiny) + attention logits, fused per node
// ---------------------------------------------------------------------------
__global__ void l2_node(const float* __restrict__ hfc, const float* __restrict__ W2,
                        const float* __restrict__ att_s, const float* __restrict__ att_d,
                        float* __restrict__ h2, float* __restrict__ a_s, float* __restrict__ a_d)
{
    const int n = blockIdx.x * blockDim.x + threadIdx.x;
    if (n >= NN) return;
    float acc[HC2] = {};
    const float* xp = hfc + (long long)n * HIDD;
    for (int k = 0; k < HIDD; ++k) {
        const float xv = xp[k];
#pragma unroll
        for (int j = 0; j < HC2; ++j) acc[j] += xv * W2[k * HC2 + j];
    }
#pragma unroll
    for (int j = 0; j < HC2; ++j) h2[(long long)n * HC2 + j] = acc[j];
#pragma unroll
    for (int hh = 0; hh < HEADS; ++hh) {
        a_s[n * HEADS + hh] = acc[hh * OUT_C] * att_s[hh * OUT_C]
                            + acc[hh * OUT_C + 1] * att_s[hh * OUT_C + 1];
        a_d[n * HEADS + hh] = acc[hh * OUT_C] * att_d[hh * OUT_C]
                            + acc[hh * OUT_C + 1] * att_d[hh * OUT_C + 1];
    }
}

// ---------------------------------------------------------------------------
// Layer-2 aggregation (fused denom): one thread per edge, 3 heads x 2 channels,
// unnormalized weights; finalize divides by den.
// ---------------------------------------------------------------------------
__global__ void edge_agg2(const int* __restrict__ ei, const float* __restrict__ h2,
                          const float* __restrict__ a_s, const float* __restrict__ a_d,
                          const unsigned* __restrict__ emax, float* __restrict__ den,
                          float* __restrict__ agg)
{
    const long long e = (long long)blockIdx.x * blockDim.x + threadIdx.x;
    if (e >= EN) return;
    int s, d; edge_sd(ei, e, s, d);
#pragma unroll
    for (int hh = 0; hh < HEADS; ++hh) {
        float el = a_s[s * HEADS + hh] + a_d[d * HEADS + hh];
        el = (el > 0.f) ? el : NEG_SLOPE * el;
        const float ex = __expf(el - funkey(emax[d * HEADS + hh]));
        atomicAdd(&den[d * HEADS + hh], ex);
        atomicAdd(&agg[(long long)d * HC2 + hh * OUT_C],     h2[(long long)s * HC2 + hh * OUT_C]     * ex);
        atomicAdd(&agg[(long long)d * HC2 + hh * OUT_C + 1], h2[(long long)s * HC2 + hh * OUT_C + 1] * ex);
    }
}

// ---------------------------------------------------------------------------
// Finalize: normalize per (dst,head), mean over heads, + bias, 2-way softmax
// ---------------------------------------------------------------------------
__global__ void finalize(const float* __restrict__ agg2, const float* __restrict__ den,
                         const float* __restrict__ b2, float* __restrict__ out)
{
    const int n = blockIdx.x * blockDim.x + threadIdx.x;
    if (n >= NN) return;
    const float* a = agg2 + (long long)n * HC2;
    const float i0 = 1.f / den[n * HEADS + 0];
    const float i1 = 1.f / den[n * HEADS + 1];
    const float i2 = 1.f / den[n * HEADS + 2];
    const float v0 = (a[0] * i0 + a[2] * i1 + a[4] * i2) * (1.f / 3.f) + b2[0];
    const float v1 = (a[1] * i0 + a[3] * i1 + a[5] * i2) * (1.f / 3.f) + b2[1];
    const float m  = fmaxf(v0, v1);
    const float e0 = __expf(v0 - m), e1 = __expf(v1 - m);
    const float inv = 1.f / (e0 + e1);
    out[n * 2 + 0] = e0 * inv;
    out[n * 2 + 1] = e1 * inv;
}

__global__ void fill_u32(unsigned* __restrict__ p, unsigned v, int n)
{
    const int i = blockIdx.x * blockDim.x + threadIdx.x;
    if (i < n) p[i] = v;
}

// ---------------------------------------------------------------------------
extern "C" void kernel_launch(void* const* d_in, const int* in_sizes, int n_in,
                              void* d_out, int out_size, void* d_ws, size_t ws_size,
                              hipStream_t stream)
{
    (void)in_sizes; (void)n_in; (void)out_size; (void)ws_size;

    const float* x        = (const float*)d_in[0];
    const int*   ei       = (const int*)  d_in[1];
    const float* g1_W     = (const float*)d_in[2];
    const float* g1_as    = (const float*)d_in[3];
    const float* g1_ad    = (const float*)d_in[4];
    const float* g1_b     = (const float*)d_in[5];
    const float* fc_W     = (const float*)d_in[6];
    const float* fc_b     = (const float*)d_in[7];
    const float* g2_W     = (const float*)d_in[8];
    const float* g2_as    = (const float*)d_in[9];
    const float* g2_ad    = (const float*)d_in[10];
    const float* g2_b     = (const float*)d_in[11];
    float* out = (float*)d_out;

    // ---- workspace layout (all from d_ws; ~194 MB) ----
    char* ws = (char*)d_ws;
    size_t off = 0;
    auto alloc = [&](size_t bytes) -> void* {
        void* p = ws + off;
        off = (off + bytes + 255) & ~(size_t)255;
        return p;
    };
    float*    h1     = (float*)   alloc((size_t)NN * HC1 * 4);   // 76.8 MB
    float*    a1s    = (float*)   alloc((size_t)NN * HEADS * 4);
    float*    a1d    = (float*)   alloc((size_t)NN * HEADS * 4);
    unsigned* emax1  = (unsigned*)alloc((size_t)NN * HEADS * 4);
    float*    den1   = (float*)   alloc((size_t)NN * HEADS * 4);
    float*    agg1   = (float*)   alloc((size_t)NN * HC1 * 4);   // 76.8 MB
    float*    hfc    = (float*)   alloc((size_t)NN * HIDD * 4);  // 25.6 MB
    float*    h2     = (float*)   alloc((size_t)NN * HC2 * 4);
    float*    a2s    = (float*)   alloc((size_t)NN * HEADS * 4);
    float*    a2d    = (float*)   alloc((size_t)NN * HEADS * 4);
    unsigned* emax2  = (unsigned*)alloc((size_t)NN * HEADS * 4);
    float*    den2   = (float*)   alloc((size_t)NN * HEADS * 4);
    float*    agg2   = (float*)   alloc((size_t)NN * HC2 * 4);

    // ---- per-call init (graph-capture safe) ----
    hipMemsetAsync(den1, 0, (size_t)NN * HEADS * 4, stream);
    hipMemsetAsync(agg1, 0, (size_t)NN * HC1   * 4, stream);
    hipMemsetAsync(den2, 0, (size_t)NN * HEADS * 4, stream);
    hipMemsetAsync(agg2, 0, (size_t)NN * HC2   * 4, stream);
    const unsigned NEG_INF_KEY = 0x007FFFFFu;                 // fkey(-inf)
    fill_u32<<<(NN * HEADS + 255) / 256, 256, 0, stream>>>(emax1, NEG_INF_KEY, NN * HEADS);
    fill_u32<<<(NN * HEADS + 255) / 256, 256, 0, stream>>>(emax2, NEG_INF_KEY, NN * HEADS);

    const int tiles_m  = NN / 16;                  // 6250
    const int gx       = (tiles_m + 7) / 8;        // 8 waves (row tiles) per block

    // ---- Layer 1: h1 = x @ W1  (WMMA f32; g1_b folded into FC A-load) ----
    wmma_gemm_f32<F_INN, false, false>
        <<<dim3(gx, HC1 / 16), 256, 0, stream>>>(x, nullptr, g1_W, nullptr, h1, NN, HC1);

    att_logits1<<<(NN * HEADS + 255) / 256, 256, 0, stream>>>(h1, g1_as, g1_ad, a1s, a1d);

    const long long TE = (long long)EN * HEADS;    // 5.1M edge-head items
    edge_max <<<(int)((TE + 255) / 256), 256, 0, stream>>>(ei, a1s, a1d, emax1);
    edge_agg1<<<(int)((TE + 7) / 8), 256, 0, stream>>>(ei, h1, a1s, a1d, emax1, den1, agg1);
    norm1    <<<(NN * HC1 + 255) / 256, 256, 0, stream>>>(agg1, den1);

    // ---- FC: hfc = relu((agg1 + g1_b) @ fc_W + fc_b) ----
    wmma_gemm_f32<HC1, true, true>
        <<<dim3(gx, HIDD / 16), 256, 0, stream>>>(agg1, g1_b, fc_W, fc_b, hfc, NN, HIDD);

    // ---- Layer 2 ----
    l2_node<<<(NN + 255) / 256, 256, 0, stream>>>(hfc, g2_W, g2_as, g2_ad, h2, a2s, a2d);
    edge_max <<<(int)((TE + 255) / 256), 256, 0, stream>>>(ei, a2s, a2d, emax2);
    edge_agg2<<<(EN + 255) / 256, 256, 0, stream>>>(ei, h2, a2s, a2d, emax2, den2, agg2);

    finalize<<<(NN + 255) / 256, 256, 0, stream>>>(agg2, den2, g2_b, out);
}